// EDTAttention_18708877541696
// MI455X (gfx1250) — compile-verified
//
#include <hip/hip_runtime.h>

// MI455X / gfx1250, wave32. fp32 end-to-end using V_WMMA_F32_16X16X4_F32.
// Round 3: fix async global->LDS builtin argument types (int4 AS1 / AS3 ptrs).

typedef __attribute__((ext_vector_type(2))) float v2f;
typedef __attribute__((ext_vector_type(4))) float v4f;
typedef __attribute__((ext_vector_type(8))) float v8f;
typedef __attribute__((ext_vector_type(4))) int v4i;

#if __has_builtin(__builtin_amdgcn_global_load_async_to_lds_b128)
#define EDT_ASYNC_LDS 1
typedef __attribute__((address_space(1))) v4i edt_gv4i;  // global int4
typedef __attribute__((address_space(3))) v4i edt_sv4i;  // shared int4
#else
#define EDT_ASYNC_LDS 0
#endif

__device__ __forceinline__ v8f wmma_k4(v2f a, v2f b, v8f c) {
  // D = A(16x4) * B(4x16) + C(16x16), all f32
  return __builtin_amdgcn_wmma_f32_16x16x4_f32(false, a, false, b, (short)0, c,
                                               false, false);
}

__device__ __forceinline__ float wave_sum(float v) {
#pragma unroll
  for (int m = 16; m >= 1; m >>= 1) v += __shfl_xor(v, m, 32);
  return v;
}
__device__ __forceinline__ float wave_max(float v) {
#pragma unroll
  for (int m = 16; m >= 1; m >>= 1) v = fmaxf(v, __shfl_xor(v, m, 32));
  return v;
}

// Shapes: B=8, S=256, N=256, C=64, NH=8, D=2048. Y = qkv out: (524288, 256).

// ---------------------------------------------------------------------------
// K1: Y[m,f] = sum_c xf[m,c] * qkv_w[f,c] + qkv_b[f].  M=524288, K=64, N=256.
// Wave tile: 16 x 64 (4 v8f accumulators). 131072 waves.
// ---------------------------------------------------------------------------
__global__ __launch_bounds__(256) void edt_qkv(const float* __restrict__ xf,
                                               const float* __restrict__ w,
                                               const float* __restrict__ bias,
                                               float* __restrict__ Y) {
  const int tid = threadIdx.x;
  const int lane = tid & 31, row = lane & 15, kh = lane >> 4;
  const long wid = (long)blockIdx.x * 8 + (tid >> 5);
  const long m0 = (wid >> 2) * 16;
  const int n0 = (int)(wid & 3) * 64;

  v8f acc[4] = {};
  const float* arow = xf + (m0 + row) * 64;
#pragma unroll
  for (int k0 = 0; k0 < 64; k0 += 4) {
    const int kk = k0 + 2 * kh;
    v2f a = *(const v2f*)(arow + kk);
#pragma unroll
    for (int j = 0; j < 4; ++j) {
      const int col = n0 + j * 16 + row;
      v2f b = *(const v2f*)(w + col * 64 + kk);  // B[k,n] = qkv_w[n,k]
      acc[j] = wmma_k4(a, b, acc[j]);
    }
  }
#pragma unroll
  for (int j = 0; j < 4; ++j) {
    const int col = n0 + j * 16 + row;
    const float bb = bias[col];
#pragma unroll
    for (int i = 0; i < 8; ++i) {
      const long m = m0 + i + 8 * kh;
      Y[m * 256 + col] = acc[j][i] + bb;
    }
  }
}

// ---------------------------------------------------------------------------
// K2: per (b,h,s): 1/max(||q||,eps), 1/max(||k||,eps). One wave per row pair.
// ---------------------------------------------------------------------------
__global__ __launch_bounds__(256) void edt_norm(const float* __restrict__ Y,
                                                float* __restrict__ invq,
                                                float* __restrict__ invk) {
  const int tid = threadIdx.x, lane = tid & 31;
  const int wid = blockIdx.x * 8 + (tid >> 5);  // 0..16383 == bh*256 + s
  const int bh = wid >> 8, s = wid & 255;
  const int b = bh >> 3, h = bh & 7;
  const float* q = Y + (long)(b * 65536 + s * 256 + h * 8) * 256;
  const float* k = q + 64 * 256;
  float sq = 0.f, sk = 0.f;
#pragma unroll 4
  for (int i = 0; i < 16; ++i) {
    v4f a = *(const v4f*)(q + i * 128 + lane * 4);
    sq += a[0] * a[0] + a[1] * a[1] + a[2] * a[2] + a[3] * a[3];
    v4f c = *(const v4f*)(k + i * 128 + lane * 4);
    sk += c[0] * c[0] + c[1] * c[1] + c[2] * c[2] + c[3] * c[3];
  }
  sq = wave_sum(sq);
  sk = wave_sum(sk);
  if (lane == 0) {
    invq[wid] = 1.0f / fmaxf(sqrtf(sq), 1e-12f);
    invk[wid] = 1.0f / fmaxf(sqrtf(sk), 1e-12f);
  }
}

// ---------------------------------------------------------------------------
// K3: sim[bh,s,t] = invq[s]*invk[t] * sum_d Q[s,d]K[t,d].  K=2048.
// Wave tile 32x64 (8 accumulators): 8 WMMA per 6 b64 loads. 2048 waves.
// ---------------------------------------------------------------------------
__global__ __launch_bounds__(256) void edt_sim(const float* __restrict__ Y,
                                               const float* __restrict__ invq,
                                               const float* __restrict__ invk,
                                               float* __restrict__ sim) {
  const int tid = threadIdx.x, lane = tid & 31, row = lane & 15, kh = lane >> 4;
  const int wid = blockIdx.x * 8 + (tid >> 5);  // 0..2047
  const int bh = wid >> 5, t5 = wid & 31;
  const int b = bh >> 3, h = bh & 7;
  const int m0 = (t5 >> 2) * 32;  // 8 m-tiles of 32
  const int t0 = (t5 & 3) * 64;   // 4 n-tiles of 64

  const float* qb = Y + (long)(b * 65536 + h * 8) * 256;  // + s*65536 + d
  const float* kbp = qb + 64 * 256;
  const float* qr0 = qb + (long)(m0 + row) * 65536;
  const float* qr1 = qr0 + 16L * 65536;
  const float* kr0 = kbp + (long)(t0 + row) * 65536;
  const float* kr1 = kr0 + 16L * 65536;
  const float* kr2 = kr0 + 32L * 65536;
  const float* kr3 = kr0 + 48L * 65536;

  v8f acc[2][4] = {};
#pragma unroll 2
  for (int k0 = 0; k0 < 2048; k0 += 4) {
    const int kk = k0 + 2 * kh;
    v2f a0 = *(const v2f*)(qr0 + kk);
    v2f a1 = *(const v2f*)(qr1 + kk);
    v2f b0 = *(const v2f*)(kr0 + kk);
    v2f b1 = *(const v2f*)(kr1 + kk);
    v2f b2 = *(const v2f*)(kr2 + kk);
    v2f b3 = *(const v2f*)(kr3 + kk);
    acc[0][0] = wmma_k4(a0, b0, acc[0][0]);
    acc[0][1] = wmma_k4(a0, b1, acc[0][1]);
    acc[0][2] = wmma_k4(a0, b2, acc[0][2]);
    acc[0][3] = wmma_k4(a0, b3, acc[0][3]);
    acc[1][0] = wmma_k4(a1, b0, acc[1][0]);
    acc[1][1] = wmma_k4(a1, b1, acc[1][1]);
    acc[1][2] = wmma_k4(a1, b2, acc[1][2]);
    acc[1][3] = wmma_k4(a1, b3, acc[1][3]);
  }

  const float* iq = invq + bh * 256;
  const float* ik = invk + bh * 256;
  float ikv[4];
#pragma unroll
  for (int j = 0; j < 4; ++j) ikv[j] = ik[t0 + j * 16 + row];
  float* simb = sim + (long)bh * 65536;
#pragma unroll
  for (int mi = 0; mi < 2; ++mi) {
#pragma unroll
    for (int i = 0; i < 8; ++i) {
      const int s = m0 + mi * 16 + i + 8 * kh;
      const float qs = iq[s];
#pragma unroll
      for (int j = 0; j < 4; ++j)
        simb[(long)s * 256 + t0 + j * 16 + row] = acc[mi][j][i] * qs * ikv[j];
    }
  }
}

// ---------------------------------------------------------------------------
// K4: dual softmax. One wave per (bh,s) row of 256.
// ---------------------------------------------------------------------------
__global__ __launch_bounds__(256) void edt_softmax(const float* __restrict__ sim,
                                                   const float* __restrict__ t1,
                                                   const float* __restrict__ t2,
                                                   float* __restrict__ PH,
                                                   float* __restrict__ PV) {
  const int tid = threadIdx.x, lane = tid & 31;
  const int wid = blockIdx.x * 8 + (tid >> 5);  // bh*256 + s
  const int h = (wid >> 8) & 7;
  const float th = t1[h], tv = t2[h];
  const float* rowp = sim + (long)wid * 256;
  v4f x0 = *(const v4f*)(rowp + lane * 8);
  v4f x1 = *(const v4f*)(rowp + lane * 8 + 4);
  float vh[8], vv[8];
#pragma unroll
  for (int i = 0; i < 4; ++i) {
    vh[i] = x0[i] * th; vh[4 + i] = x1[i] * th;
    vv[i] = x0[i] * tv; vv[4 + i] = x1[i] * tv;
  }
  float mh = vh[0], mv = vv[0];
#pragma unroll
  for (int i = 1; i < 8; ++i) { mh = fmaxf(mh, vh[i]); mv = fmaxf(mv, vv[i]); }
  mh = wave_max(mh); mv = wave_max(mv);
  float sh = 0.f, sv = 0.f;
#pragma unroll
  for (int i = 0; i < 8; ++i) {
    vh[i] = __expf(vh[i] - mh); sh += vh[i];
    vv[i] = __expf(vv[i] - mv); sv += vv[i];
  }
  sh = 1.0f / wave_sum(sh);
  sv = 1.0f / wave_sum(sv);
  v4f oh0, oh1, ov0, ov1;
#pragma unroll
  for (int i = 0; i < 4; ++i) {
    oh0[i] = vh[i] * sh; oh1[i] = vh[4 + i] * sh;
    ov0[i] = vv[i] * sv; ov1[i] = vv[4 + i] * sv;
  }
  *(v4f*)(PH + (long)wid * 256 + lane * 8) = oh0;
  *(v4f*)(PH + (long)wid * 256 + lane * 8 + 4) = oh1;
  *(v4f*)(PV + (long)wid * 256 + lane * 8) = ov0;
  *(v4f*)(PV + (long)wid * 256 + lane * 8 + 4) = ov1;
}

// ---------------------------------------------------------------------------
// K5: dense gather of v_V:  Vv[bh,t,d] = Y[b, (h*32+d/64)*256 + 192 +
//     (t/32)*8 + (t%32)/4, (t%4)*64 + d%64].  64-float contiguous chunks.
// ---------------------------------------------------------------------------
__global__ __launch_bounds__(256) void edt_vgather(const float* __restrict__ Y,
                                                   float* __restrict__ Vv) {
  const int tid = threadIdx.x, lane = tid & 31;
  const int wid = blockIdx.x * 8 + (tid >> 5);  // bh*256 + t
  const int bh = wid >> 8, t = wid & 255;
  const int b = bh >> 3, h = bh & 7;
  const int colbase = (t & 3) * 64;
  const long rbase = (long)b * 65536 + 192 + (t >> 5) * 8 + ((t & 31) >> 2);
  float* dst = Vv + (long)wid * 2048;
#pragma unroll 4
  for (int g = 0; g < 32; ++g) {
    const float* src = Y + (rbase + (h * 32 + g) * 256) * 256 + colbase;
    *(v2f*)(dst + g * 64 + lane * 2) = *(const v2f*)(src + lane * 2);
  }
}

// ---------------------------------------------------------------------------
// K6: X = P_H @ V_H + P_V @ V_V per (b,h). M=256, N=2048, K=256.
// Workgroup (8 waves) computes 128x128; V_H/V_V k-slabs (32x128) staged in LDS
// via async global->LDS (shared by all 8 waves). Wave tile 16x128 (8 v8f).
// LDS rows padded to 136 floats so the two half-wave K-phases use disjoint
// bank groups. 2048 blocks.
// ---------------------------------------------------------------------------
#define AV_LDS_STRIDE 136  // 128 + 8 pad; 2*136 % 64 == 16 -> no bank overlap

__global__ __launch_bounds__(256) void edt_av(const float* __restrict__ Y,
                                              const float* __restrict__ PH,
                                              const float* __restrict__ PV,
                                              const float* __restrict__ Vv,
                                              float* __restrict__ X) {
  __shared__ float sVH[32 * AV_LDS_STRIDE];
  __shared__ float sVV[32 * AV_LDS_STRIDE];

  const int tid = threadIdx.x, lane = tid & 31, row = lane & 15, kh = lane >> 4;
  const int bh = blockIdx.x >> 5;        // 32 blocks per (b,h)
  const int blk = blockIdx.x & 31;
  const int b = bh >> 3, h = bh & 7;
  const int M0 = (blk >> 4) * 128;       // token block
  const int N0 = (blk & 15) * 128;       // feature block
  const int m0 = M0 + (tid >> 5) * 16;   // this wave's 16 token rows

  const float* aHrow = PH + (long)(bh * 256 + m0 + row) * 256;
  const float* aVrow = PV + (long)(bh * 256 + m0 + row) * 256;
  const float* vhb = Y + (long)(b * 65536 + 128 + h * 8) * 256;  // + t*65536 + d
  const float* vvb = Vv + (long)bh * 524288;                     // + t*2048  + d

  v8f acc[8] = {};

  for (int kb = 0; kb < 256; kb += 32) {
    if (kb) __syncthreads();  // previous slab fully consumed
    // ---- stage V_H / V_V k-slab (32 x 128 each) into LDS ----
#pragma unroll
    for (int i = 0; i < 4; ++i) {
      const int f4 = tid + i * 256;          // 1024 float4 per slab
      const int k = f4 >> 5, n4 = f4 & 31;   // k row, float4 within row
      const float* gH = vhb + (long)(kb + k) * 65536 + N0 + n4 * 4;
      const float* gV = vvb + (long)(kb + k) * 2048 + N0 + n4 * 4;
      float* lH = sVH + k * AV_LDS_STRIDE + n4 * 4;
      float* lV = sVV + k * AV_LDS_STRIDE + n4 * 4;
#if EDT_ASYNC_LDS
      __builtin_amdgcn_global_load_async_to_lds_b128((edt_gv4i*)gH,
                                                     (edt_sv4i*)lH, 0, 0);
      __builtin_amdgcn_global_load_async_to_lds_b128((edt_gv4i*)gV,
                                                     (edt_sv4i*)lV, 0, 0);
#else
      *(v4f*)lH = *(const v4f*)gH;
      *(v4f*)lV = *(const v4f*)gV;
#endif
    }
#if EDT_ASYNC_LDS
#if __has_builtin(__builtin_amdgcn_s_wait_asynccnt)
    __builtin_amdgcn_s_wait_asynccnt(0);
#else
    asm volatile("s_wait_asynccnt 0" ::: "memory");
#endif
#endif
    __syncthreads();

    // ---- compute: 8 k-steps of 4, 16 WMMA each ----
#pragma unroll 2
    for (int k0 = 0; k0 < 32; k0 += 4) {
      const int kk = k0 + 2 * kh;
      v2f ah = *(const v2f*)(aHrow + kb + kk);
      v2f av = *(const v2f*)(aVrow + kb + kk);
#pragma unroll
      for (int j = 0; j < 8; ++j) {
        const int col = j * 16 + row;
        v2f bH, bV;
        bH[0] = sVH[kk * AV_LDS_STRIDE + col];
        bH[1] = sVH[(kk + 1) * AV_LDS_STRIDE + col];
        bV[0] = sVV[kk * AV_LDS_STRIDE + col];
        bV[1] = sVV[(kk + 1) * AV_LDS_STRIDE + col];
        acc[j] = wmma_k4(ah, bH, acc[j]);
        acc[j] = wmma_k4(av, bV, acc[j]);
      }
    }
  }

  // X[(b*65536 + s*256 + h*32)*64 + d], d in [0,2048) contiguous per (s)
  float* xb = X + (long)(b * 65536 + h * 32) * 64;
#pragma unroll
  for (int j = 0; j < 8; ++j) {
    const int col = N0 + j * 16 + row;
#pragma unroll
    for (int i = 0; i < 8; ++i) {
      const int s = m0 + i + 8 * kh;
      xb[(long)s * 16384 + col] = acc[j][i];
    }
  }
}

// ---------------------------------------------------------------------------
// K7: out = X @ proj_w.T + proj_b.  M=524288, K=64, N=64. Wave tile 16x64.
// ---------------------------------------------------------------------------
__global__ __launch_bounds__(256) void edt_proj(const float* __restrict__ X,
                                                const float* __restrict__ w,
                                                const float* __restrict__ bias,
                                                float* __restrict__ out) {
  const int tid = threadIdx.x;
  const int lane = tid & 31, row = lane & 15, kh = lane >> 4;
  const long wid = (long)blockIdx.x * 8 + (tid >> 5);
  const long m0 = wid * 16;

  v8f acc[4] = {};
  const float* arow = X + (m0 + row) * 64;
#pragma unroll
  for (int k0 = 0; k0 < 64; k0 += 4) {
    const int kk = k0 + 2 * kh;
    v2f a = *(const v2f*)(arow + kk);
#pragma unroll
    for (int j = 0; j < 4; ++j) {
      const int col = j * 16 + row;
      v2f b = *(const v2f*)(w + col * 64 + kk);
      acc[j] = wmma_k4(a, b, acc[j]);
    }
  }
#pragma unroll
  for (int j = 0; j < 4; ++j) {
    const int col = j * 16 + row;
    const float bb = bias[col];
#pragma unroll
    for (int i = 0; i < 8; ++i) {
      const long m = m0 + i + 8 * kh;
      out[m * 64 + col] = acc[j][i] + bb;
    }
  }
}

// ---------------------------------------------------------------------------
// Workspace layout (floats)
// ---------------------------------------------------------------------------
static constexpr size_t NY = 134217728ULL;   // Y  : (524288, 256)
static constexpr size_t NINV = 16384ULL;     // invq / invk : (64, 256)
static constexpr size_t NSIM = 4194304ULL;   // sim / PH / PV : (64, 256, 256)
static constexpr size_t NV = 33554432ULL;    // Vv : (64, 256, 2048)
static constexpr size_t OFF_Y = 0;
static constexpr size_t OFF_IQ = OFF_Y + NY;
static constexpr size_t OFF_IK = OFF_IQ + NINV;
static constexpr size_t OFF_SIM = OFF_IK + NINV;
static constexpr size_t OFF_PH = OFF_SIM + NSIM;
static constexpr size_t OFF_PV = OFF_PH + NSIM;
static constexpr size_t OFF_VV = OFF_PV + NSIM;
static constexpr size_t OFF_X = OFF_VV + NV;

extern "C" void kernel_launch(void* const* d_in, const int* in_sizes, int n_in,
                              void* d_out, int out_size, void* d_ws,
                              size_t ws_size, hipStream_t stream) {
  (void)in_sizes; (void)n_in; (void)out_size; (void)ws_size;
  const float* x = (const float*)d_in[0];
  const float* qkv_w = (const float*)d_in[1];
  const float* qkv_b = (const float*)d_in[2];
  const float* proj_w = (const float*)d_in[3];
  const float* proj_b = (const float*)d_in[4];
  const float* temp1 = (const float*)d_in[5];
  const float* temp2 = (const float*)d_in[6];
  float* out = (float*)d_out;

  float* ws = (float*)d_ws;
  float* Y = ws + OFF_Y;
  float* invq = ws + OFF_IQ;
  float* invk = ws + OFF_IK;
  float* sim = ws + OFF_SIM;
  float* PH = ws + OFF_PH;
  float* PV = ws + OFF_PV;
  float* Vv = ws + OFF_VV;
  float* X = ws + OFF_X;

  const dim3 blk(256);
  edt_qkv<<<dim3(16384), blk, 0, stream>>>(x, qkv_w, qkv_b, Y);
  edt_norm<<<dim3(2048), blk, 0, stream>>>(Y, invq, invk);
  edt_sim<<<dim3(256), blk, 0, stream>>>(Y, invq, invk, sim);
  edt_softmax<<<dim3(2048), blk, 0, stream>>>(sim, temp1, temp2, PH, PV);
  edt_vgather<<<dim3(2048), blk, 0, stream>>>(Y, Vv);
  edt_av<<<dim3(2048), blk, 0, stream>>>(Y, PH, PV, Vv, X);
  edt_proj<<<dim3(4096), blk, 0, stream>>>(X, proj_w, proj_b, out);
}